// AnomalyAttention_88648124989931
// MI455X (gfx1250) — compile-verified
//
#include <hip/hip_runtime.h>
#include <math.h>

typedef __attribute__((ext_vector_type(2))) float v2f;
typedef __attribute__((ext_vector_type(8))) float v8f;

#define WMMA_F32(a,b,c) __builtin_amdgcn_wmma_f32_16x16x4_f32(false,(a),false,(b),(short)0,(c),false,false)

constexpr int Bc = 2, Lc = 2048, Dc = 256, Hc = 8, HDc = 32;

// ---------------------------------------------------------------------------
// Kernel 1: Q/K/V projection.  out = x @ W^T + b.
// Q,V stored [B,H,L,hd];  K stored TRANSPOSED [B,H,hd,L] so the QK^T
// B-fragment loads in k_attn are lane-coalesced.
// grid (128, 4, 3), block 256 (8 waves, each owns one 16x16 output tile)
// ---------------------------------------------------------------------------
__global__ __launch_bounds__(256) void k_qkv(
    const float* __restrict__ x,
    const float* __restrict__ Wq, const float* __restrict__ bq,
    const float* __restrict__ Wk, const float* __restrict__ bk,
    const float* __restrict__ Wv, const float* __restrict__ bv,
    float* __restrict__ Q, float* __restrict__ Kt, float* __restrict__ V)
{
    const int z = blockIdx.z;
    const float* W    = (z == 0) ? Wq : ((z == 1) ? Wk : Wv);
    const float* bias = (z == 0) ? bq : ((z == 1) ? bk : bv);
    float*       Out  = (z == 0) ? Q  : ((z == 1) ? Kt : V);

    const int tid  = threadIdx.x;
    const int wv   = tid >> 5, lane = tid & 31;
    const int lm   = lane & 15, half = lane >> 4;
    const int mi   = wv >> 2, ni = wv & 3;
    const int mbase = blockIdx.x * 32 + mi * 16;
    const int nbase = blockIdx.y * 64 + ni * 16;

    const float* xr = x + (size_t)(mbase + lm) * Dc;
    const float* wr = W + (size_t)(nbase + lm) * Dc;   // B[k][n] = W[n][k]

    v8f acc = {};
    #pragma unroll 4
    for (int k = 0; k < Dc; k += 4) {
        const int kA = k + (half ? 2 : 0);
        v2f a, bf;
        a.x  = xr[kA]; a.y  = xr[kA + 1];
        bf.x = wr[kA]; bf.y = wr[kA + 1];
        acc = WMMA_F32(a, bf, acc);
    }
    #pragma unroll
    for (int r = 0; r < 8; r++) {
        const int row = mbase + r + 8 * half;      // global token row in [0, B*L)
        const int col = nbase + lm;                // output feature in [0, D)
        const float val = acc[r] + bias[col];
        const int bb = row >> 11, pos = row & (Lc - 1);
        const int hh = col >> 5,  dd  = col & (HDc - 1);
        if (z == 1)   // K: [B,H,hd,L]
            Out[(((size_t)(bb * Hc + hh)) * HDc + dd) * Lc + pos] = val;
        else          // Q,V: [B,H,L,hd]
            Out[(((size_t)(bb * Hc + hh)) * Lc + pos) * HDc + dd] = val;
    }
}

// ---------------------------------------------------------------------------
// Kernel 2: prior normalizers.  denomInv[h*L+i] = 1/(sum_j exp(-(i-j)^2*s_h)+1e-6)
// ---------------------------------------------------------------------------
__global__ __launch_bounds__(256) void k_prior(const float* __restrict__ u,
                                               float* __restrict__ denomInv)
{
    const int idx = blockIdx.x * 256 + threadIdx.x;   // [0, H*L)
    const int h = idx >> 11, i = idx & (Lc - 1);
    const float uu = u[h];
    const float sh = 1.0f / (2.0f * (uu * uu + 1e-6f));
    float sum = 0.0f;
    for (int j = 0; j < Lc; j++) {
        const float d = (float)(i - j);
        sum += __expf(-d * d * sh);
    }
    denomInv[idx] = 1.0f / (sum + 1e-6f);
}

__global__ void k_zero(float* __restrict__ p, int n)
{
    const int i = blockIdx.x * 256 + threadIdx.x;
    if (i < n) p[i] = 0.0f;
}

// ---------------------------------------------------------------------------
// Kernel 3: fused attention.  One workgroup per (b, h, 32 query rows).
// Full 32x2048 score strip lives in LDS (CDNA5: 320KB/WGP).
// grid (64, 8, 2), block 256
// ---------------------------------------------------------------------------
__global__ __launch_bounds__(256) void k_attn(
    const float* __restrict__ Q, const float* __restrict__ Kt,
    const float* __restrict__ V, const float* __restrict__ u,
    const float* __restrict__ denomInv,
    float* __restrict__ ctx, float* __restrict__ disc)
{
    __shared__ float sS[32][Lc + 1];      // +1 pad: conflict-free row & column access
    __shared__ float sRed[4 * 256];       // AV k-split reduction buffer
    __shared__ float sDen[32];            // prior normalizers for this q strip

    const int tid = threadIdx.x, wv = tid >> 5, lane = tid & 31;
    const int lm = lane & 15, half = lane >> 4;
    const int qt = blockIdx.x, h = blockIdx.y, b = blockIdx.z;
    const int qbase = qt * 32;
    const size_t bh = (size_t)(b * Hc + h);
    const float* Qp  = Q  + bh * Lc * HDc;
    const float* Ktp = Kt + bh * HDc * Lc;     // [hd][L]
    const float* Vp  = V  + bh * Lc * HDc;

    if (tid < 32) sDen[tid] = denomInv[h * Lc + qbase + tid];
    const float uu = u[h];
    const float sh = 1.0f / (2.0f * (uu * uu + 1e-6f));

    // ---- Phase 1: S = (Q K^T) * 1/sqrt(hd) -> LDS ----
    const int smi = wv & 1;    // q sub-tile (0/1)
    const int ks  = wv >> 1;   // one of 4 key tiles per iteration
    const int ro  = half ? 2 : 0;
    float qa0[8], qa1[8];      // hoisted Q A-fragments (hd = 32 => 8 K-steps)
    {
        const float* qrow = Qp + (size_t)(qbase + smi * 16 + lm) * HDc;
        #pragma unroll
        for (int t = 0; t < 8; t++) {
            const int kA = t * 4 + ro;
            qa0[t] = qrow[kA]; qa1[t] = qrow[kA + 1];
        }
    }
    const float scale = 0.17677669529663689f;  // 1/sqrt(32)
    for (int kt = 0; kt < 128; kt += 4) {
        const int colbase = (kt + ks) * 16;
        // B[k][n] = K[n][k] = Kt[k][n]: lane-coalesced along n
        const float* kcol = Ktp + colbase + lm;
        v8f acc = {};
        #pragma unroll
        for (int t = 0; t < 8; t++) {
            const int kA = t * 4 + ro;
            v2f a, bf;
            a.x  = qa0[t];                  a.y  = qa1[t];
            bf.x = kcol[(size_t)kA * Lc];   bf.y = kcol[(size_t)(kA + 1) * Lc];
            acc = WMMA_F32(a, bf, acc);
        }
        #pragma unroll
        for (int r = 0; r < 8; r++)
            sS[smi * 16 + r + 8 * half][colbase + lm] = acc[r] * scale;
    }
    __syncthreads();

    // ---- Phase 2: softmax per row (8 lanes / row, shfl_xor width-8) ----
    {
        const int row = tid >> 3, sub = tid & 7;
        const int j0 = sub * 256;
        float m = -3.0e38f;
        for (int j = 0; j < 256; j++) m = fmaxf(m, sS[row][j0 + j]);
        m = fmaxf(m, __shfl_xor(m, 1, 8));
        m = fmaxf(m, __shfl_xor(m, 2, 8));
        m = fmaxf(m, __shfl_xor(m, 4, 8));
        float s = 0.0f;
        for (int j = 0; j < 256; j++) {
            const float p = __expf(sS[row][j0 + j] - m);
            sS[row][j0 + j] = p;
            s += p;
        }
        s += __shfl_xor(s, 1, 8);
        s += __shfl_xor(s, 2, 8);
        s += __shfl_xor(s, 4, 8);
        const float inv = 1.0f / s;
        for (int j = 0; j < 256; j++) sS[row][j0 + j] *= inv;
    }
    __syncthreads();

    // ---- Phase 3: discrepancy (reference reduces over H and the QUERY axis,
    //      so accumulate per-key-column sums) ----
    for (int it = 0; it < 8; it++) {
        const int j = it * 256 + tid;
        float cs = 0.0f;
        #pragma unroll 4
        for (int r = 0; r < 32; r++) {
            const float d = (float)(qbase + r - j);
            const float prior = __expf(-d * d * sh) * sDen[r];
            cs += fabsf(sS[r][j] - prior);
        }
        atomicAdd(disc + (size_t)b * Lc + j, cs);
    }

    // ---- Phase 4: out = attn @ V  (4 output tiles x 2-wave k-split) ----
    const int tile = wv & 3, khalf = wv >> 2;
    const int omi = tile >> 1, oni = tile & 1;
    v8f acc = {};
    const int k0 = khalf * 1024;
    for (int k = k0; k < k0 + 1024; k += 4) {
        v2f a, bf;
        a.x = sS[omi * 16 + lm][k + ro];
        a.y = sS[omi * 16 + lm][k + ro + 1];
        const float* vr = Vp + (size_t)(k + ro) * HDc + oni * 16 + lm;  // lane-coalesced
        bf.x = vr[0]; bf.y = vr[HDc];
        acc = WMMA_F32(a, bf, acc);
    }
    __syncthreads();
    if (khalf == 1) {
        #pragma unroll
        for (int r = 0; r < 8; r++)
            sRed[tile * 256 + (r + 8 * half) * 16 + lm] = acc[r];
    }
    __syncthreads();
    if (khalf == 0) {
        #pragma unroll
        for (int r = 0; r < 8; r++) {
            const float val = acc[r] + sRed[tile * 256 + (r + 8 * half) * 16 + lm];
            const int row = qbase + omi * 16 + r + 8 * half;
            const int col = h * HDc + oni * 16 + lm;
            ctx[((size_t)b * Lc + row) * Dc + col] = val;   // [B,L,D]
        }
    }
}

// ---------------------------------------------------------------------------
// Kernel 4: output projection  out = ctx @ Wo^T + bo.  grid (128,4), block 256
// ---------------------------------------------------------------------------
__global__ __launch_bounds__(256) void k_out(
    const float* __restrict__ ctx, const float* __restrict__ Wo,
    const float* __restrict__ bo, float* __restrict__ out)
{
    const int tid = threadIdx.x, wv = tid >> 5, lane = tid & 31;
    const int lm = lane & 15, half = lane >> 4;
    const int mi = wv >> 2, ni = wv & 3;
    const int mbase = blockIdx.x * 32 + mi * 16;
    const int nbase = blockIdx.y * 64 + ni * 16;

    const float* xr = ctx + (size_t)(mbase + lm) * Dc;
    const float* wr = Wo  + (size_t)(nbase + lm) * Dc;
    v8f acc = {};
    #pragma unroll 4
    for (int k = 0; k < Dc; k += 4) {
        const int kA = k + (half ? 2 : 0);
        v2f a, bf;
        a.x  = xr[kA]; a.y  = xr[kA + 1];
        bf.x = wr[kA]; bf.y = wr[kA + 1];
        acc = WMMA_F32(a, bf, acc);
    }
    #pragma unroll
    for (int r = 0; r < 8; r++) {
        const int row = mbase + r + 8 * half, col = nbase + lm;
        out[(size_t)row * Dc + col] = acc[r] + bo[col];
    }
}

__global__ void k_disc(const float* __restrict__ dws, float* __restrict__ dout)
{
    const int i = blockIdx.x * 256 + threadIdx.x;
    if (i < Bc * Lc) dout[i] = dws[i] * (1.0f / (float)(Hc * Lc));  // mean over (H, Lq)
}

// ---------------------------------------------------------------------------
extern "C" void kernel_launch(void* const* d_in, const int* in_sizes, int n_in,
                              void* d_out, int out_size, void* d_ws, size_t ws_size,
                              hipStream_t stream)
{
    const float* x  = (const float*)d_in[0];
    const float* Wq = (const float*)d_in[1];
    const float* bq = (const float*)d_in[2];
    const float* Wk = (const float*)d_in[3];
    const float* bk = (const float*)d_in[4];
    const float* Wv = (const float*)d_in[5];
    const float* bv = (const float*)d_in[6];
    const float* u  = (const float*)d_in[7];
    const float* Wo = (const float*)d_in[8];
    const float* bo = (const float*)d_in[9];

    float* out      = (float*)d_out;                      // [B,L,D]
    float* disc_out = out + (size_t)Bc * Lc * Dc;         // [B,L]

    float* ws       = (float*)d_ws;
    float* Q        = ws;                                 // 1,048,576 f32
    float* Kt       = Q + (size_t)Bc * Hc * Lc * HDc;     // transposed [B,H,hd,L]
    float* V        = Kt + (size_t)Bc * Hc * Lc * HDc;
    float* ctx      = V + (size_t)Bc * Hc * Lc * HDc;     // 1,048,576 f32
    float* denomInv = ctx + (size_t)Bc * Lc * Dc;         // 16,384 f32
    float* dws      = denomInv + Hc * Lc;                 // 4,096 f32

    k_qkv <<<dim3(128, 4, 3), 256, 0, stream>>>(x, Wq, bq, Wk, bk, Wv, bv, Q, Kt, V);
    k_prior<<<dim3(64),       256, 0, stream>>>(u, denomInv);
    k_zero <<<dim3(16),       256, 0, stream>>>(dws, Bc * Lc);
    k_attn <<<dim3(64, 8, 2), 256, 0, stream>>>(Q, Kt, V, u, denomInv, ctx, dws);
    k_out  <<<dim3(128, 4),   256, 0, stream>>>(ctx, Wo, bo, out);
    k_disc <<<dim3(16),       256, 0, stream>>>(dws, disc_out);
}